// RecurrentHedgeModel_66683662238302
// MI455X (gfx1250) — compile-verified
//
#include <hip/hip_runtime.h>

typedef __attribute__((ext_vector_type(2))) float v2f;
typedef __attribute__((ext_vector_type(8))) float v8f;

namespace {
constexpr int kSteps         = 30;
constexpr int kWavesPerBlock = 8;
constexpr int kBatchPerWave  = 16;
constexpr int kBatchPerBlock = kWavesPerBlock * kBatchPerWave;  // 128
}  // namespace

// Single-instruction ReLU. fmaxf/fmed3 both lower to canonicalize(max(x,x))
// + max on this toolchain; raw v_max_num_f32 with inline-constant 0 avoids
// the extra op and also keeps the SLP vectorizer from inserting pairing movs.
static __device__ __forceinline__ float relu1(float x) {
  float r;
  asm("v_max_num_f32 %0, 0, %1" : "=v"(r) : "v"(x));
  return r;
}

// Half-wave swap (lane n <-> lane n^16): ds_swizzle SWAPX16 immediate
// (group-of-32 mode: xor=0x10, or=0, and=0x1f -> 0x401F).
static __device__ __forceinline__ float swap16(float x) {
  return __int_as_float(__builtin_amdgcn_ds_swizzle(__float_as_int(x), 0x401F));
}

// One wave32 handles a 16-element batch tile for all 30 recurrent steps.
// Layer 1 ([B,3]@[3,32]+b1) runs on the matrix pipe as two
// v_wmma_f32_16x16x4_f32 (A = W1^T slice, B = inp^T, C = b1 broadcast).
// Layer 2 (relu + dot with W2) runs on VALU with one ds_swizzle swap.
__global__ __launch_bounds__(kWavesPerBlock * 32) void
hedge_wmma_kernel(const float* __restrict__ S,
                  const float* __restrict__ W1,
                  const float* __restrict__ b1,
                  const float* __restrict__ W2,
                  const float* __restrict__ b2,
                  const float* __restrict__ a_init,
                  float* __restrict__ out,
                  int Btot) {
  const int tid  = threadIdx.x;
  const int lane = tid & 31;
  const int wave = tid >> 5;
  const int base = (blockIdx.x * kWavesPerBlock + wave) * kBatchPerWave;
  if (base >= Btot) return;  // wave-uniform: EXEC stays all-ones for live waves

  const int  n  = lane & 15;  // batch column (B/D) or hid row (A)
  const bool lo = lane < 16;
  const int  bat = base + n;

  // A matrices: W1^T padded to K=4 (K3 = 0).
  // f32 16x4 A layout: lanes 0-15 (M=lane): v0=K0, v1=K1 ;
  //                    lanes 16-31 (M=lane-16): v0=K2, v1=K3.
  v2f a_lo, a_hi;
  a_lo.x = lo ? W1[n * 3 + 0] : W1[n * 3 + 2];
  a_lo.y = lo ? W1[n * 3 + 1] : 0.0f;
  a_hi.x = lo ? W1[(n + 16) * 3 + 0] : W1[(n + 16) * 3 + 2];
  a_hi.y = lo ? W1[(n + 16) * 3 + 1] : 0.0f;

  // C matrices: b1 broadcast along batch columns.
  // f32 16x16 C/D layout: lanes 0-15: VGPR v = row M=v ; lanes 16-31: M=v+8.
  v8f c_lo, c_hi;
#pragma unroll
  for (int v = 0; v < 8; ++v) {
    c_lo[v] = lo ? b1[v]      : b1[v + 8];
    c_hi[v] = lo ? b1[v + 16] : b1[v + 24];
  }

  // W2 slices matching the D layout (per half-wave).
  float w2lo[8], w2hi[8];
#pragma unroll
  for (int v = 0; v < 8; ++v) {
    w2lo[v] = lo ? W2[v]      : W2[v + 8];
    w2hi[v] = lo ? W2[v + 16] : W2[v + 24];
  }

  // Fold b2 into the reduction: seed one accumulator with b2/2 in BOTH
  // half-waves; after part + swap16(part) it contributes exactly b2 once.
  // Keeps the scalar b2 load out of the recurrent loop entirely.
  const float halfb2 = 0.5f * b2[0];

  // Preload this tile's input series (streamed once -> non-temporal).
  float s_arr[kSteps];
#pragma unroll
  for (int t = 0; t < kSteps; ++t)
    s_arr[t] = __builtin_nontemporal_load(&S[bat * kSteps + t]);

  float h = 0.0f;       // EMA state
  float d = a_init[0];  // d_prev, replicated in both half-waves

  // Paired stores: lanes 0-15 write step t-1, lanes 16-31 write step t,
  // so every other step issues one full-width 128B store.
  const int outbase = bat * kSteps - (lo ? 1 : 0);
  float dkeep = 0.0f;

#pragma unroll
  for (int t = 0; t < kSteps; ++t) {
    // B matrix = inp^T (4x16): K rows {S_t, d_prev, h, 0}.
    // f32 4x16 B layout: lanes 0-15 (N=lane): v0=K0, v1=K1 ;
    //                    lanes 16-31 (N=lane-16): v0=K2, v1=K3.
    v2f bm;
    bm.x = lo ? s_arr[t] : h;
    bm.y = lo ? d : 0.0f;

    v8f hid_lo = __builtin_amdgcn_wmma_f32_16x16x4_f32(
        false, a_lo, false, bm, (short)0, c_lo, false, false);
    v8f hid_hi = __builtin_amdgcn_wmma_f32_16x16x4_f32(
        false, a_hi, false, bm, (short)0, c_hi, false, false);

    // Layer 2: d = W2 . relu(hid) + b2, 4 independent FMA chains.
    float acc0 = halfb2, acc1 = 0.0f, acc2 = 0.0f, acc3 = 0.0f;
#pragma unroll
    for (int v = 0; v < 4; ++v) {
      acc0 = fmaf(relu1(hid_lo[v]),     w2lo[v],     acc0);
      acc1 = fmaf(relu1(hid_lo[v + 4]), w2lo[v + 4], acc1);
      acc2 = fmaf(relu1(hid_hi[v]),     w2hi[v],     acc2);
      acc3 = fmaf(relu1(hid_hi[v + 4]), w2hi[v + 4], acc3);
    }
    const float part = (acc0 + acc1) + (acc2 + acc3);
    const float dval = part + swap16(part);

    h = fmaf(0.2f, dval, 0.8f * h);
    d = dval;

    if (t & 1) {
      const float vstore = lo ? dkeep : dval;
      __builtin_nontemporal_store(vstore, &out[outbase + t]);
    } else {
      dkeep = dval;
    }
  }
}

extern "C" void kernel_launch(void* const* d_in, const int* in_sizes, int n_in,
                              void* d_out, int out_size, void* d_ws, size_t ws_size,
                              hipStream_t stream) {
  const float* S      = (const float*)d_in[0];
  const float* W1     = (const float*)d_in[1];
  const float* b1     = (const float*)d_in[2];
  const float* W2     = (const float*)d_in[3];
  const float* b2     = (const float*)d_in[4];
  const float* a_init = (const float*)d_in[5];
  float* out = (float*)d_out;

  const int Btot   = in_sizes[0] / kSteps;
  const int blocks = (Btot + kBatchPerBlock - 1) / kBatchPerBlock;
  hedge_wmma_kernel<<<blocks, kWavesPerBlock * 32, 0, stream>>>(
      S, W1, b1, W2, b2, a_init, out, Btot);
}